// EquivariantProductBasisBlock_34445637714613
// MI455X (gfx1250) — compile-verified
//
#include <hip/hip_runtime.h>

#define NNODES 512
#define CH 64
#define NF 9
#define NTRI 165
#define NPAIR 45
#define KPAD 224      // 165 + 45 + 9 = 219, padded to multiple of 4
#define NPADC 96      // 64 + 16 + 4 = 84 cols, padded to 6 WMMA N-tiles
#define MONO_STRIDE 228   // 228 % 64 = 36 -> conflict-free row striping

typedef __attribute__((ext_vector_type(2))) float v2f;
typedef __attribute__((ext_vector_type(8))) float v8f;

// ---- compile-time tables of sorted monomials + permutation multiplicities ----
struct Tab {
  unsigned char ti[NTRI], tj[NTRI], tk[NTRI]; float tm[NTRI];
  unsigned char pi[NPAIR], pj[NPAIR];         float pm[NPAIR];
};
constexpr Tab makeTab() {
  Tab T{};
  int t = 0;
  for (int i = 0; i < NF; ++i)
    for (int j = i; j < NF; ++j)
      for (int k = j; k < NF; ++k) {
        T.ti[t] = (unsigned char)i; T.tj[t] = (unsigned char)j; T.tk[t] = (unsigned char)k;
        float m = 6.f;
        if (i == j && j == k) m = 1.f;
        else if (i == j || j == k) m = 3.f;
        T.tm[t] = m; ++t;
      }
  int p = 0;
  for (int i = 0; i < NF; ++i)
    for (int j = i; j < NF; ++j) {
      T.pi[p] = (unsigned char)i; T.pj[p] = (unsigned char)j;
      T.pm[p] = (i == j) ? 1.f : 2.f; ++p;
    }
  return T;
}
__constant__ Tab g_tab = makeTab();

// ---- kernel 1: pack multiplicity-weighted U tensors into Cpack[KPAD][NPADC] ----
// column map: [0..15] U3_0(m) | [16..63] U3_1(d*16+m) | [64..67] U2_0(m)
//             [68..79] U2_1(d*4+m) | [80] U1_0 | [81..83] U1_1(d) | rest 0
__global__ void pack_coeff(const float* __restrict__ U3_0, const float* __restrict__ U2_0,
                           const float* __restrict__ U1_0, const float* __restrict__ U3_1,
                           const float* __restrict__ U2_1, const float* __restrict__ U1_1,
                           float* __restrict__ Cpack) {
  int t = blockIdx.x * blockDim.x + threadIdx.x;
  if (t >= KPAD * NPADC) return;
  int row = t / NPADC, col = t % NPADC;
  float v = 0.f;
  if (row < NTRI) {
    int i = g_tab.ti[row], j = g_tab.tj[row], k = g_tab.tk[row];
    float m = g_tab.tm[row];
    if (col < 16) {
      v = m * U3_0[((i * 9 + j) * 9 + k) * 16 + col];
    } else if (col < 64) {
      int cc = col - 16, d = cc >> 4, mm = cc & 15;
      v = m * U3_1[(((d * 9 + i) * 9 + j) * 9 + k) * 16 + mm];
    }
  } else if (row < NTRI + NPAIR) {
    int p = row - NTRI;
    int i = g_tab.pi[p], j = g_tab.pj[p];
    float m = g_tab.pm[p];
    if (col >= 64 && col < 68) {
      v = m * U2_0[(i * 9 + j) * 4 + (col - 64)];
    } else if (col >= 68 && col < 80) {
      int cc = col - 68, d = cc >> 2, mm = cc & 3;
      v = m * U2_1[((d * 9 + i) * 9 + j) * 4 + mm];
    }
  } else if (row < NTRI + NPAIR + NF) {
    int i = row - (NTRI + NPAIR);
    if (col == 80) v = U1_0[i];
    else if (col >= 81 && col < 84) v = U1_1[(col - 81) * 9 + i];
  }
  Cpack[t] = v;
}

// ---- kernel 2: fused monomial build -> WMMA GEMM -> species-weight reduction ----
// 4 waves per block; wave w owns M-tile (16 consecutive (b,c) rows).
__global__ __launch_bounds__(128)
void mono_gemm_reduce(const float* __restrict__ feats,   // [32768][9]
                      const int*   __restrict__ specie,  // [512]
                      const float* __restrict__ Cpack,   // [KPAD][NPADC]
                      const float* __restrict__ w3_0, const float* __restrict__ w2_0,
                      const float* __restrict__ w1_0, const float* __restrict__ w3_1,
                      const float* __restrict__ w2_1, const float* __restrict__ w1_1,
                      float* __restrict__ F)             // [32768][4]
{
  __shared__ float ldsm[4][16 * MONO_STRIDE];
  const int lane  = threadIdx.x & 31;
  const int w     = threadIdx.x >> 5;
  const int mtile = blockIdx.x * 4 + w;
  float* M = &ldsm[w][0];

  // phase 1: polynomial basis (monomials) into LDS
  {
    const int r   = lane & 15;
    const int row = mtile * 16 + r;
    float x[NF];
#pragma unroll
    for (int i = 0; i < NF; ++i) x[i] = feats[row * NF + i];
    const int t0 = (lane < 16) ? 0 : 112;
    const int t1 = (lane < 16) ? 112 : KPAD;
    for (int t = t0; t < t1; ++t) {
      float v;
      if (t < NTRI)                  v = x[g_tab.ti[t]] * x[g_tab.tj[t]] * x[g_tab.tk[t]];
      else if (t < NTRI + NPAIR)     { int p = t - NTRI; v = x[g_tab.pi[p]] * x[g_tab.pj[p]]; }
      else if (t < NTRI + NPAIR + NF) v = x[t - (NTRI + NPAIR)];
      else                            v = 0.f;
      M[r * MONO_STRIDE + t] = v;
    }
  }
  __syncthreads();

  // phase 2: D[16][96] = Mono[16][224] x Cpack[224][96] via f32 WMMA 16x16x4
  v8f acc[6] = {};
  const int arow = lane & 15;
  const int koff = (lane < 16) ? 0 : 2;
  for (int kk = 0; kk < KPAD; kk += 4) {
    v2f a;
    a.x = M[arow * MONO_STRIDE + kk + koff];
    a.y = M[arow * MONO_STRIDE + kk + koff + 1];
#pragma unroll
    for (int nt = 0; nt < 6; ++nt) {
      v2f b;
      const float* bp = Cpack + (kk + koff) * NPADC + nt * 16 + arow;
      b.x = bp[0];
      b.y = bp[NPADC];
      acc[nt] = __builtin_amdgcn_wmma_f32_16x16x4_f32(false, a, false, b,
                                                      (short)0, acc[nt], false, false);
    }
  }
  __syncthreads();

  // phase 3: spill D tile into (reused) LDS region
  {
    const int n     = lane & 15;
    const int mbase = (lane < 16) ? 0 : 8;
#pragma unroll
    for (int nt = 0; nt < 6; ++nt)
#pragma unroll
      for (int v = 0; v < 8; ++v)
        M[(mbase + v) * MONO_STRIDE + nt * 16 + n] = acc[nt][v];
  }
  __syncthreads();

  // phase 4: contract with per-species weights (21 MACs per (row, d))
  for (int task = lane; task < 64; task += 32) {
    const int r   = task >> 2;
    const int d   = task & 3;
    const int row = mtile * 16 + r;
    const int b   = row >> 6;
    const int c   = row & 63;
    const int s   = specie[b];
    const float* D = &M[r * MONO_STRIDE];
    float f = 0.f;
    if (d == 0) {
#pragma unroll
      for (int m = 0; m < 16; ++m) f += w3_0[(s * 16 + m) * 64 + c] * D[m];
#pragma unroll
      for (int m = 0; m < 4; ++m)  f += w2_0[(s * 4 + m) * 64 + c] * D[64 + m];
      f += w1_0[s * 64 + c] * D[80];
    } else {
      const int dd = d - 1;
#pragma unroll
      for (int m = 0; m < 16; ++m) f += w3_1[(s * 16 + m) * 64 + c] * D[16 + dd * 16 + m];
#pragma unroll
      for (int m = 0; m < 4; ++m)  f += w2_1[(s * 4 + m) * 64 + c] * D[68 + dd * 4 + m];
      f += w1_1[s * 64 + c] * D[81 + dd];
    }
    F[row * 4 + d] = f;
  }
}

// ---- kernel 3: final e3nn linear, 4 GEMMs [512,64]x[64,64] via f32 WMMA ----
__global__ __launch_bounds__(32)
void final_linear(const float* __restrict__ F,      // [512][64][4]
                  const float* __restrict__ Wlin0,  // [64][64]
                  const float* __restrict__ Wlin1,  // [64][64]
                  float* __restrict__ out)          // [512][256]
{
  const int mtile = blockIdx.x;          // 0..31 (node tiles)
  const int d     = blockIdx.y;          // 0..3 (output irrep component)
  const int lane  = threadIdx.x & 31;
  const float* W  = (d == 0) ? Wlin0 : Wlin1;
  const int arow  = lane & 15;
  const int koff  = (lane < 16) ? 0 : 2;
  const int b0    = mtile * 16;

  v8f acc[4] = {};
  for (int kk = 0; kk < 64; kk += 4) {
    v2f a;
    a.x = F[((b0 + arow) * 64 + kk + koff) * 4 + d];
    a.y = F[((b0 + arow) * 64 + kk + koff + 1) * 4 + d];
#pragma unroll
    for (int nt = 0; nt < 4; ++nt) {
      v2f b;
      const float* bp = W + (kk + koff) * 64 + nt * 16 + arow;
      b.x = bp[0];
      b.y = bp[64];
      acc[nt] = __builtin_amdgcn_wmma_f32_16x16x4_f32(false, a, false, b,
                                                      (short)0, acc[nt], false, false);
    }
  }

  const int n16   = lane & 15;
  const int mbase = (lane < 16) ? 0 : 8;
#pragma unroll
  for (int nt = 0; nt < 4; ++nt)
#pragma unroll
    for (int v = 0; v < 8; ++v) {
      const int b   = b0 + mbase + v;
      const int n   = nt * 16 + n16;
      const float val = acc[nt][v] * 0.125f;   // 1/sqrt(CHANNELS)
      if (d == 0) out[b * 256 + n] = val;
      else        out[b * 256 + 64 + n * 3 + (d - 1)] = val;
    }
}

extern "C" void kernel_launch(void* const* d_in, const int* in_sizes, int n_in,
                              void* d_out, int out_size, void* d_ws, size_t ws_size,
                              hipStream_t stream) {
  const float* feats = (const float*)d_in[0];
  const int*   spec  = (const int*)d_in[1];
  const float* U3_0  = (const float*)d_in[2];
  const float* U2_0  = (const float*)d_in[3];
  const float* U1_0  = (const float*)d_in[4];
  const float* w3_0  = (const float*)d_in[5];
  const float* w2_0  = (const float*)d_in[6];
  const float* w1_0  = (const float*)d_in[7];
  const float* U3_1  = (const float*)d_in[8];
  const float* U2_1  = (const float*)d_in[9];
  const float* U1_1  = (const float*)d_in[10];
  const float* w3_1  = (const float*)d_in[11];
  const float* w2_1  = (const float*)d_in[12];
  const float* w1_1  = (const float*)d_in[13];
  const float* Wl0   = (const float*)d_in[14];
  const float* Wl1   = (const float*)d_in[15];

  float* Cpack = (float*)d_ws;                     // 224*96 floats
  float* F     = (float*)d_ws + KPAD * NPADC;      // 32768*4 floats

  pack_coeff<<<(KPAD * NPADC + 255) / 256, 256, 0, stream>>>(
      U3_0, U2_0, U1_0, U3_1, U2_1, U1_1, Cpack);
  mono_gemm_reduce<<<512, 128, 0, stream>>>(
      feats, spec, Cpack, w3_0, w2_0, w1_0, w3_1, w2_1, w1_1, F);
  final_linear<<<dim3(32, 4), 32, 0, stream>>>(F, Wl0, Wl1, (float*)d_out);
}